// MoEGate_60705067762031
// MI455X (gfx1250) — compile-verified
//
#include <hip/hip_runtime.h>
#include <hip/hip_bf16.h>
#include <stdint.h>

// Problem constants (from reference): B=4, T=4096, C=2048, E=8, K=8
#define NTOK  16384      // B*T
#define CDIM  2048
#define EEXP  8
#define KTOP  8
#define NK    (NTOK * KTOP)   // 131072

#define WAVES   8
#define KSLICE  (CDIM / WAVES)   // 256 channels per wave (split-K)
#define CHUNK   32               // floats per row per staged chunk (128 B)
#define NCHUNK  (KSLICE / CHUNK) // 8
#define ROWP    36               // padded LDS row stride (floats) -> bank-conflict-free

typedef float v2f __attribute__((ext_vector_type(2)));
typedef float v8f __attribute__((ext_vector_type(8)));

// Block = 256 threads = 8 waves; all 8 waves cooperate on ONE 16-token tile,
// split-K over C (each wave reduces 256 channels). Grid = 16384/16 = 1024
// blocks -> 8192 waves (enough in-flight work to saturate 23.3 TB/s HBM).
// A-tile staged global->LDS with async b128 copies (coalesced 512B bursts),
// double-buffered via s_wait_asynccnt; WMMA = V_WMMA_F32_16X16X4_F32.
__global__ __launch_bounds__(256) void moe_gate_kernel(
    const float* __restrict__ x,        // [NTOK, CDIM]
    const uint8_t* __restrict__ mask,   // [NTOK] (bool)
    const float* __restrict__ gw,       // [EEXP, CDIM]
    const float* __restrict__ gb,       // [EEXP]
    const float* __restrict__ eb,       // [EEXP]
    float* __restrict__ outF,           // [NK idx][NK probs][1 maxvio]
    float* __restrict__ counts)         // [EEXP] in d_ws
{
    __shared__ float sx[2][WAVES][16][ROWP]; // double-buffered A staging (36 KB)
    __shared__ float sD[WAVES][16][EEXP];    // per-wave partial D (4 KB)
    __shared__ float sCount[EEXP];

    const int tid  = threadIdx.x;
    const int wave = tid >> 5;
    const int lane = tid & 31;
    const int n    = lane & 15;        // N index (expert column; 8..15 pad)
    const int h    = lane >> 4;        // half-wave selector (K sub-pair)

    if (tid < EEXP) sCount[tid] = 0.0f;

    const int m0 = blockIdx.x * 16;    // 16-token tile shared by all waves
    const int kb = wave * KSLICE;      // this wave's K slice

    // ---- async-copy lane mapping: instr j moves rows 4j..4j+3, 16 B/lane ----
    const int arow = (lane >> 3);          // 0..3 row within group of 4
    const int acol = (lane & 7) * 4;       // float offset within 32-float chunk
    const float* gsrc[4];
    uint32_t ldsa[2][4];
    #pragma unroll
    for (int j = 0; j < 4; ++j) {
        gsrc[j] = x + (size_t)(m0 + 4 * j + arow) * CDIM + kb + acol;
        #pragma unroll
        for (int b = 0; b < 2; ++b)
            ldsa[b][j] = (uint32_t)(uintptr_t)&sx[b][wave][4 * j + arow][acol];
    }

    // B fragments direct from global (64 KB weight matrix is L2/L0 resident);
    // pad columns n>=8 are zeroed by bs.
    const float* wb = gw + (size_t)(n & (EEXP - 1)) * CDIM + kb + 2 * h;
    const float bs = (n < EEXP) ? 1.0f : 0.0f;

    v8f acc0 = {}, acc1 = {}, acc2 = {}, acc3 = {};

    // prime buffer 0 with chunk 0
    #pragma unroll
    for (int j = 0; j < 4; ++j) {
        asm volatile("global_load_async_to_lds_b128 %0, %1, off"
                     :: "v"(ldsa[0][j]), "v"(gsrc[j]) : "memory");
    }

    for (int c = 0; c < NCHUNK; ++c) {
        const int buf = c & 1;
        if (c + 1 < NCHUNK) {
            // issue next chunk into the other buffer, then retire current chunk
            #pragma unroll
            for (int j = 0; j < 4; ++j) {
                const float* g = gsrc[j] + (c + 1) * CHUNK;
                asm volatile("global_load_async_to_lds_b128 %0, %1, off"
                             :: "v"(ldsa[buf ^ 1][j]), "v"(g) : "memory");
            }
            asm volatile("s_wait_asynccnt 4" ::: "memory");
        } else {
            asm volatile("s_wait_asynccnt 0" ::: "memory");
        }

        const float* wc = wb + c * CHUNK;
        // 8 K-steps of 4; A fragment from LDS (conflict-free b64 reads)
        #pragma unroll
        for (int i = 0; i < 8; ++i) {
            v2f a = *(const v2f*)&sx[buf][wave][n][4 * i + 2 * h];
            v2f b = *(const v2f*)(wc + 4 * i) * bs;
            // 8 args: (neg_a, A, neg_b, B, c_mod, C, reuse_a, reuse_b)
            if ((i & 3) == 0)
                acc0 = __builtin_amdgcn_wmma_f32_16x16x4_f32(false, a, false, b, (short)0, acc0, false, false);
            else if ((i & 3) == 1)
                acc1 = __builtin_amdgcn_wmma_f32_16x16x4_f32(false, a, false, b, (short)0, acc1, false, false);
            else if ((i & 3) == 2)
                acc2 = __builtin_amdgcn_wmma_f32_16x16x4_f32(false, a, false, b, (short)0, acc2, false, false);
            else
                acc3 = __builtin_amdgcn_wmma_f32_16x16x4_f32(false, a, false, b, (short)0, acc3, false, false);
        }
    }

    v8f acc = (acc0 + acc1) + (acc2 + acc3);

    // D layout: lane (n, h), VGPR r -> D[M = r + 8h][N = n]. Stash partials.
    if (n < EEXP) {
        #pragma unroll
        for (int r = 0; r < 8; ++r) {
            sD[wave][r + 8 * h][n] = acc[r];
        }
    }
    __syncthreads();

    // Threads 0-15: each owns one token; sum the 8 split-K partials, then
    // full stable descending argsort of the 8 logits (K == E == 8).
    if (tid < 16) {
        const int tok = m0 + tid;
        float key[EEXP];  // sort key: gate_output + expert_bias
        float pr[EEXP];   // sigmoid(gate_output)
        int   idx[EEXP];
        #pragma unroll
        for (int e = 0; e < EEXP; ++e) {
            float go = 0.0f;
            #pragma unroll
            for (int w = 0; w < WAVES; ++w) go += sD[w][tid][e];
            go += gb[e];
            pr[e]  = 1.0f / (1.0f + __expf(-go));
            key[e] = go + eb[e];
            idx[e] = e;
        }
        // Stable bubble sort, fully unrolled (register-only: constant indices)
        #pragma unroll
        for (int a = 0; a < EEXP - 1; ++a) {
            #pragma unroll
            for (int b = 0; b < EEXP - 1 - a; ++b) {
                if (key[b] < key[b + 1]) {
                    float tk = key[b]; key[b] = key[b + 1]; key[b + 1] = tk;
                    float tp = pr[b];  pr[b]  = pr[b + 1];  pr[b + 1]  = tp;
                    int   ti = idx[b]; idx[b] = idx[b + 1]; idx[b + 1] = ti;
                }
            }
        }
        float psum = 0.0f;
        #pragma unroll
        for (int j = 0; j < KTOP; ++j) psum += pr[j];
        const float rinv = 1.0f / psum;

        const float m = mask[tok] ? 1.0f : 0.0f;
        #pragma unroll
        for (int j = 0; j < KTOP; ++j) {
            outF[(size_t)tok * KTOP + j]      = (float)idx[j];
            outF[NK + (size_t)tok * KTOP + j] = pr[j] * rinv;
            atomicAdd(&sCount[idx[j]], m);
        }
    }
    __syncthreads();
    if (tid < EEXP) atomicAdd(&counts[tid], sCount[tid]);
}

__global__ void moe_zero_counts(float* counts) {
    if (threadIdx.x < EEXP) counts[threadIdx.x] = 0.0f;
}

__global__ void moe_finalize(const float* __restrict__ counts, float* __restrict__ outF) {
    if (threadIdx.x == 0) {
        float sum = 0.0f, mx = -1.0f;
        #pragma unroll
        for (int e = 0; e < EEXP; ++e) {
            float c = counts[e];
            sum += c;
            mx = fmaxf(mx, c);
        }
        float avg = sum / (float)EEXP;
        outF[2 * NK] = (mx - avg) / (avg + 1e-5f);
    }
}

extern "C" void kernel_launch(void* const* d_in, const int* in_sizes, int n_in,
                              void* d_out, int out_size, void* d_ws, size_t ws_size,
                              hipStream_t stream) {
    const float*   x    = (const float*)d_in[0];
    const uint8_t* mask = (const uint8_t*)d_in[1];
    const float*   gw   = (const float*)d_in[2];
    const float*   gb   = (const float*)d_in[3];
    const float*   eb   = (const float*)d_in[4];
    float* outF   = (float*)d_out;
    float* counts = (float*)d_ws;

    moe_zero_counts<<<1, 32, 0, stream>>>(counts);
    moe_gate_kernel<<<NTOK / 16, 256, 0, stream>>>(x, mask, gw, gb, eb, outF, counts);
    moe_finalize<<<1, 32, 0, stream>>>(counts, outF);
}